// Attention_61194694034287
// MI455X (gfx1250) — compile-verified
//
#include <hip/hip_runtime.h>

// Attention with softmax over the HEAD axis (reference quirk: softmax(axis=1))
// and scale 1/sqrt(H) (reference quirk: d = k.shape[1] == H == 8).
// B=2, H=8, S=2048, D=64. fp32 throughout, fp32 WMMA (V_WMMA_F32_16X16X4_F32).

#define B_  2
#define H_  8
#define S_  2048
#define D_  64

typedef __attribute__((ext_vector_type(2))) float v2f;
typedef __attribute__((ext_vector_type(8))) float v8f;

__device__ __forceinline__ v8f wmma_f32(v2f a, v2f b, v8f c) {
    // 8 args: (neg_a, A, neg_b, B, c_mod, C, reuse_a, reuse_b)
    return __builtin_amdgcn_wmma_f32_16x16x4_f32(false, a, false, b,
                                                 (short)0, c, false, false);
}

// ---------------------------------------------------------------------------
// Kernel 1: scores = mask ? (Q @ K^T) / sqrt(8) : -1e11   (pre-softmax)
// One wave -> 16 (query) x 64 (key) strip: 4 accumulators, 16 K-steps, 64 WMMAs.
// ---------------------------------------------------------------------------
__global__ __launch_bounds__(256) void qk_scores_kernel(
    const float* __restrict__ q, const float* __restrict__ k,
    const int* __restrict__ mask, float* __restrict__ att)
{
    const int lane = threadIdx.x & 31;
    const int wave = threadIdx.x >> 5;
    const int tile = blockIdx.x * 8 + wave;        // 0 .. 65535

    const int nChunk = tile & 31;                  // 64-wide key chunk (32 per row)
    const int mTile  = (tile >> 5) & 127;          // 16-wide query tile
    const int bh     = tile >> 12;                 // 0 .. 15
    const int b      = bh / H_;

    const float* Q = q + (size_t)bh * S_ * D_ + (size_t)mTile * 16 * D_;
    const float* K = k + (size_t)bh * S_ * D_ + (size_t)nChunk * 64 * D_;

    const int mn   = lane & 15;          // M for A-frag, N for B-frag
    const int koff = (lane >> 4) * 2;    // K-pair selected by half-wave

    v8f acc0 = {}, acc1 = {}, acc2 = {}, acc3 = {};

    #pragma unroll
    for (int k0 = 0; k0 < D_; k0 += 4) {
        v2f a = *(const v2f*)(Q + mn * D_ + k0 + koff);          // 8B aligned
        v2f b0 = *(const v2f*)(K + (mn +  0) * D_ + k0 + koff);
        v2f b1 = *(const v2f*)(K + (mn + 16) * D_ + k0 + koff);
        v2f b2 = *(const v2f*)(K + (mn + 32) * D_ + k0 + koff);
        v2f b3 = *(const v2f*)(K + (mn + 48) * D_ + k0 + koff);
        acc0 = wmma_f32(a, b0, acc0);
        acc1 = wmma_f32(a, b1, acc1);
        acc2 = wmma_f32(a, b2, acc2);
        acc3 = wmma_f32(a, b3, acc3);
    }

    // scale by 1/sqrt(H) (reference: d = k.shape[1] = 8), mask, store.
    const float scale = 0.35355339059327373f;      // 1/sqrt(8)
    const int q0  = mTile * 16;
    const int k0g = nChunk * 64;
    const int n      = lane & 15;
    const int mhalf  = (lane >> 4) * 8;            // C/D layout: M split by half-wave
    const int* mrow  = mask + (size_t)b * S_ * S_;
    float* out       = att  + (size_t)bh * S_ * S_;

    v8f accs[4] = {acc0, acc1, acc2, acc3};
    #pragma unroll
    for (int t = 0; t < 4; ++t) {
        #pragma unroll
        for (int r = 0; r < 8; ++r) {
            const int mm = q0 + mhalf + r;
            const int nn = k0g + t * 16 + n;
            float val = accs[t][r] * scale;
            if (mrow[(size_t)mm * S_ + nn] == 0) val = -1.0e11f;
            out[(size_t)mm * S_ + nn] = val;
        }
    }
}

// ---------------------------------------------------------------------------
// Kernel 2: softmax over the h axis (stride S*S), in place.
// One thread per (b, q, k) triple; 8 strided reads/writes, coalesced per wave.
// ---------------------------------------------------------------------------
__global__ __launch_bounds__(256) void softmax_h_kernel(float* __restrict__ att)
{
    const size_t idx = (size_t)blockIdx.x * 256 + threadIdx.x;   // b*S*S domain
    const size_t b   = idx / ((size_t)S_ * S_);
    const size_t qk  = idx % ((size_t)S_ * S_);
    float* base = att + b * H_ * (size_t)S_ * S_ + qk;

    float vals[H_];
    float mx = -3.4e38f;
    #pragma unroll
    for (int h = 0; h < H_; ++h) {
        vals[h] = base[(size_t)h * S_ * S_];
        mx = fmaxf(mx, vals[h]);
    }
    float sum = 0.0f;
    #pragma unroll
    for (int h = 0; h < H_; ++h) {
        vals[h] = __expf(vals[h] - mx);    // v_exp_f32
        sum += vals[h];
    }
    const float inv = 1.0f / sum;
    #pragma unroll
    for (int h = 0; h < H_; ++h)
        base[(size_t)h * S_ * S_] = vals[h] * inv;
}

// ---------------------------------------------------------------------------
// Kernel 3: z = att @ V.  One wave -> 16 (query) x 64 (= full D) z-tile.
// K = 2048 in steps of 4; A-fragment reused across the 4 N-tiles.
// ---------------------------------------------------------------------------
__global__ __launch_bounds__(256) void pv_kernel(
    const float* __restrict__ att, const float* __restrict__ v,
    float* __restrict__ z)
{
    const int lane = threadIdx.x & 31;
    const int wave = threadIdx.x >> 5;
    const int tile = blockIdx.x * 8 + wave;        // 0 .. 2047

    const int mTile = tile & 127;                  // 16-wide query tile
    const int bh    = tile >> 7;                   // 0 .. 15

    const float* A = att + (size_t)bh * S_ * S_ + (size_t)mTile * 16 * S_;
    const float* V = v   + (size_t)bh * S_ * D_;

    const int mn   = lane & 15;
    const int koff = (lane >> 4) * 2;

    v8f acc0 = {}, acc1 = {}, acc2 = {}, acc3 = {};

    #pragma unroll 4
    for (int kk = 0; kk < S_; kk += 4) {
        v2f a = *(const v2f*)(A + (size_t)mn * S_ + kk + koff);  // 8B aligned
        const float* Vr0 = V + (size_t)(kk + koff)     * D_ + mn;
        const float* Vr1 = V + (size_t)(kk + koff + 1) * D_ + mn;
        v2f b0, b1, b2, b3;
        b0.x = Vr0[ 0]; b0.y = Vr1[ 0];
        b1.x = Vr0[16]; b1.y = Vr1[16];
        b2.x = Vr0[32]; b2.y = Vr1[32];
        b3.x = Vr0[48]; b3.y = Vr1[48];
        acc0 = wmma_f32(a, b0, acc0);
        acc1 = wmma_f32(a, b1, acc1);
        acc2 = wmma_f32(a, b2, acc2);
        acc3 = wmma_f32(a, b3, acc3);
    }

    const int q0    = mTile * 16;
    const int n     = lane & 15;
    const int mhalf = (lane >> 4) * 8;
    float* out = z + (size_t)bh * S_ * D_;

    v8f accs[4] = {acc0, acc1, acc2, acc3};
    #pragma unroll
    for (int t = 0; t < 4; ++t) {
        #pragma unroll
        for (int r = 0; r < 8; ++r) {
            const int mm = q0 + mhalf + r;
            out[(size_t)mm * D_ + t * 16 + n] = accs[t][r];
        }
    }
}

// ---------------------------------------------------------------------------
extern "C" void kernel_launch(void* const* d_in, const int* in_sizes, int n_in,
                              void* d_out, int out_size, void* d_ws, size_t ws_size,
                              hipStream_t stream) {
    const float* q    = (const float*)d_in[0];
    const float* k    = (const float*)d_in[1];
    const float* v    = (const float*)d_in[2];
    const int*   mask = (const int*)  d_in[3];

    float* z   = (float*)d_out;                           // B*H*S*D floats
    float* att = z + (size_t)B_ * H_ * S_ * D_;           // B*H*S*S floats

    // 1) pre-softmax masked scores -> att region of d_out
    //    waves = B*H * (S/16) * (S/64) = 65536 -> 8192 blocks x 8 waves
    qk_scores_kernel<<<8192, 256, 0, stream>>>(q, k, mask, att);

    // 2) softmax over h, in place: B*S*S = 8388608 threads -> 32768 blocks
    softmax_h_kernel<<<32768, 256, 0, stream>>>(att);

    // 3) z = att @ V: waves = B*H * (S/16) = 2048 -> 256 blocks x 8 waves
    pv_kernel<<<256, 256, 0, stream>>>(att, v, z);
}